// SampleConditionalGMM_80917183856859
// MI455X (gfx1250) — compile-verified
//
#include <hip/hip_runtime.h>
#include <hip/hip_bf16.h>
#include <stdint.h>

// Problem geometry (from reference): B=2, D=H=W=192, L=32, C=3.
#define DHW_  (192*192*192)     // 7,077,888 spatial points per batch
#define HALF_ (3*DHW_)          // 21,233,664 = N/2 threefry counter offset
#define TAB_  192               // floats per table (B*L*C = 2*32*3)

#if defined(__has_builtin)
# if __has_builtin(__builtin_amdgcn_global_load_async_to_lds_b32)
#  define HAVE_ASYNC_LDS 1
# endif
#endif
#ifndef HAVE_ASYNC_LDS
# define HAVE_ASYNC_LDS 0
#endif

// Address-space-qualified pointer casts (clang prints AS1 as "__device__").
#define GLOBAL_I32(p) ((__attribute__((address_space(1))) int*)(p))
#define LDS_I32(p)    ((__attribute__((address_space(3))) int*)(p))

__device__ __forceinline__ uint32_t rotl32(uint32_t x, uint32_t r) {
  return (x << r) | (x >> (32u - r));
}

// Threefry-2x32-20 with key = (0, 42)  == jax.random.key(42)
__device__ __forceinline__ void threefry2x32_k42(uint32_t x0, uint32_t x1,
                                                 uint32_t& o0, uint32_t& o1) {
  const uint32_t ks0 = 0u;
  const uint32_t ks1 = 42u;
  const uint32_t ks2 = 0x1BD11BDAu ^ ks0 ^ ks1;
  x0 += ks0; x1 += ks1;
#define TF_R(r) { x0 += x1; x1 = rotl32(x1, (r)); x1 ^= x0; }
  TF_R(13) TF_R(15) TF_R(26) TF_R(6)
  x0 += ks1; x1 += ks2 + 1u;
  TF_R(17) TF_R(29) TF_R(16) TF_R(24)
  x0 += ks2; x1 += ks0 + 2u;
  TF_R(13) TF_R(15) TF_R(26) TF_R(6)
  x0 += ks0; x1 += ks1 + 3u;
  TF_R(17) TF_R(29) TF_R(16) TF_R(24)
  x0 += ks1; x1 += ks2 + 4u;
  TF_R(13) TF_R(15) TF_R(26) TF_R(6)
  x0 += ks2; x1 += ks0 + 5u;
#undef TF_R
  o0 = x0; o1 = x1;
}

// JAX mapping: bits -> uniform in [nextafter(-1,0), 1) -> sqrt(2)*erfinv(u)
__device__ __forceinline__ float gauss_from_bits(uint32_t bits) {
  float f = __uint_as_float((bits >> 9) | 0x3f800000u);  // [1, 2)
  const float lo   = -0.99999994f;                       // nextafter(-1, 0)
  const float span = 1.0f - lo;                          // 1.99999994f
  float u = fmaf(f, span, lo - span);                    // (f-1)*span + lo
  u = fmaxf(u, lo);
  // erfinv, single-precision (Giles 2012)
  float w = -__logf(fmaf(-u, u, 1.0f));                  // -log(1 - u*u)
  float p;
  if (w < 5.0f) {
    w -= 2.5f;
    p = 2.81022636e-08f;
    p = fmaf(p, w, 3.43273939e-07f);
    p = fmaf(p, w, -3.5233877e-06f);
    p = fmaf(p, w, -4.39150654e-06f);
    p = fmaf(p, w, 0.00021858087f);
    p = fmaf(p, w, -0.00125372503f);
    p = fmaf(p, w, -0.00417768164f);
    p = fmaf(p, w, 0.246640727f);
    p = fmaf(p, w, 1.50140941f);
  } else {
    w = sqrtf(w) - 3.0f;
    p = -0.000200214257f;
    p = fmaf(p, w, 0.000100950558f);
    p = fmaf(p, w, 0.00134934322f);
    p = fmaf(p, w, -0.00367342844f);
    p = fmaf(p, w, 0.00573950773f);
    p = fmaf(p, w, -0.0076224613f);
    p = fmaf(p, w, 0.00943887047f);
    p = fmaf(p, w, 1.00167406f);
    p = fmaf(p, w, 2.83297682f);
  }
  return 1.41421356237f * p * u;
}

// One thread = one spatial point t, handled simultaneously for batch 0 and
// batch 1 (the two batches are exactly the two threefry counter halves).
__global__ __launch_bounds__(256) void SampleConditionalGMM_kernel(
    const int* __restrict__ labels,   // [2, 192,192,192, 1] int32
    const float* __restrict__ means,  // [2, 32, 3] f32
    const float* __restrict__ stds,   // [2, 32, 3] f32
    float* __restrict__ out) {        // [2, 192,192,192, 3] f32
  // LDS table: [0..191] = means (b0 then b1), [192..383] = stds (b0 then b1)
  __shared__ float s_tab[2 * TAB_];
  const int tid = threadIdx.x;
  if (tid < TAB_) {
#if HAVE_ASYNC_LDS
    __builtin_amdgcn_global_load_async_to_lds_b32(
        GLOBAL_I32((void*)(means + tid)), LDS_I32(s_tab + tid), 0, 0);
    __builtin_amdgcn_global_load_async_to_lds_b32(
        GLOBAL_I32((void*)(stds + tid)), LDS_I32(s_tab + TAB_ + tid), 0, 0);
#else
    s_tab[tid] = means[tid];
    s_tab[TAB_ + tid] = stds[tid];
#endif
  }
#if HAVE_ASYNC_LDS
  asm volatile("s_wait_asynccnt 0" ::: "memory");
#endif
  __syncthreads();

  const int t = blockIdx.x * 256 + tid;          // spatial point (batch 0)
  const int lbl0 = labels[t];                    // batch 0 label
  const int lbl1 = labels[t + DHW_];             // batch 1 label

  const float* m0 = s_tab + lbl0 * 3;            // means, batch 0
  const float* m1 = s_tab + 96 + lbl1 * 3;       // means, batch 1
  const float* sd0 = s_tab + TAB_ + lbl0 * 3;    // stds,  batch 0
  const float* sd1 = s_tab + TAB_ + 96 + lbl1 * 3;

  float r0[3], r1[3];
  const uint32_t base = 3u * (uint32_t)t;        // element index, batch 0
#pragma unroll
  for (int c = 0; c < 3; ++c) {
    uint32_t b0, b1;
    // counter pair (i, i+N/2): b0 -> batch0 element, b1 -> batch1 element
    threefry2x32_k42(base + (uint32_t)c, base + (uint32_t)c + HALF_, b0, b1);
    r0[c] = fmaf(sd0[c], gauss_from_bits(b0), m0[c]);
    r1[c] = fmaf(sd1[c], gauss_from_bits(b1), m1[c]);
  }

  float* o0 = out + (size_t)3 * (size_t)t;
  float* o1 = out + (size_t)3 * (size_t)(t + DHW_);
  o0[0] = r0[0]; o0[1] = r0[1]; o0[2] = r0[2];
  o1[0] = r1[0]; o1[1] = r1[1]; o1[2] = r1[2];
}

extern "C" void kernel_launch(void* const* d_in, const int* in_sizes, int n_in,
                              void* d_out, int out_size, void* d_ws, size_t ws_size,
                              hipStream_t stream) {
  (void)in_sizes; (void)n_in; (void)out_size; (void)d_ws; (void)ws_size;
  const int*   labels = (const int*)d_in[0];    // [2,192,192,192,1] int32
  const float* means  = (const float*)d_in[1];  // [2,32,3] f32
  const float* stds   = (const float*)d_in[2];  // [2,32,3] f32
  float*       out    = (float*)d_out;          // [2,192,192,192,3] f32

  const int nthreads = DHW_;                    // one thread per b0 point
  dim3 grid(nthreads / 256);                    // 27648 blocks
  dim3 block(256);                              // 8 waves (wave32)
  SampleConditionalGMM_kernel<<<grid, block, 0, stream>>>(labels, means, stds, out);
}